// DMV_65807488909700
// MI455X (gfx1250) — compile-verified
//
#include <hip/hip_runtime.h>

// Inside algorithm (Eisner-style chart parser) for MI455X / gfx1250.
// The k-split logsumexp is computed in exp-space as a T x T x K f32 GEMM
// using V_WMMA_F32_16X16X4_F32 (4 waves = 4 16x16 tiles of the 32x32 chart).
// K is always zero-padded to 20 rows so the MMA chain is a compile-time
// 5x unrolled sequence: all ds_loads hoisted, one wait, 5 back-to-back WMMAs.
// Workspace layout: ct (B*NSPAN*T*V floats, ~3.3MB) then it
// (B*NSPAN*T*T*V floats, ~105MB). All launches on `stream` (graph-safe).

#define Bn 16
#define Sn 20
#define Tn 32
#define Vn 2
#define NSPAN (2 * Sn * Sn) // 800
#define KMAX 20             // max splits (19) padded to multiple of 4

typedef __attribute__((ext_vector_type(2))) float v2f;
typedef __attribute__((ext_vector_type(8))) float v8f;

__device__ __forceinline__ int sid(int l, int r, int d) {
  // matches constituent_index() enumeration order
  return 2 * (l * Sn - (l * (l - 1)) / 2 + (r - l)) + d;
}

__global__ void fill_ct_kernel(float* __restrict__ ct) {
  int idx = blockIdx.x * blockDim.x + threadIdx.x;
  if (idx < Bn * NSPAN * Tn * Vn) ct[idx] = -1.0e9f;
}

__global__ void init_diag_kernel(float* __restrict__ ct,
                                 const float* __restrict__ dec,
                                 const float* __restrict__ unary) {
  int idx = blockIdx.x * blockDim.x + threadIdx.x;
  if (idx >= Bn * Sn * Tn * Vn * 2) return;
  int d = idx % 2;            idx /= 2;
  int v = idx % Vn;           idx /= Vn;
  int t = idx % Tn;           idx /= Tn;
  int s = idx % Sn;
  int b = idx / Sn;
  // dec[b, s, t, d, v, 0] + unary[b, s, t]
  float dv = dec[(((((size_t)b * Sn + s) * Tn + t) * 2 + d) * Vn + v) * 2 + 0];
  float uv = unary[((size_t)b * Sn + s) * Tn + t];
  ct[(((size_t)b * NSPAN + sid(s, s, d)) * Tn + t) * Vn + v] = dv + uv;
}

// Incomplete spans of length L:
//   it[i,j,d][t1,t2,v] = lse_k( a[k,t1] + b[k,t2] ) + att[t1,t2] + dd
// exp-space GEMM:  M = exp(a-mA)^T * exp(b-mB)  via v_wmma_f32_16x16x4_f32
__global__ __launch_bounds__(128) void inc_kernel(
    const float* __restrict__ left_score, const float* __restrict__ right_score,
    const float* __restrict__ dec, const float* __restrict__ ct,
    float* __restrict__ it, int L) {
  const int span = blockIdx.x;
  const int b = blockIdx.y;
  const int nsp = Sn - L;
  const int d = (span >= nsp) ? 1 : 0;
  const int i = d ? (span - nsp) : span;
  const int j = i + L;
  const int K = (d == 1 && i == 0) ? 1 : L; // head-at-root restriction

  __shared__ float sA[KMAX][Tn]; // exp(a - mA), zero padded to KMAX rows
  __shared__ float sB[KMAX][Tn];
  __shared__ float mA[Tn];
  __shared__ float mB[Tn];

  const int tid = threadIdx.x;

  // stage raw split scores:  a[k,t1] = ct[i, i+k, 1][t1, v=d]
  //                          b[k,t2] = ct[i+k+1, j, 0][t2, v=1-d]
  for (int idx = tid; idx < K * Tn; idx += 128) {
    const int k = idx >> 5, t = idx & 31;
    const int kk = i + k;
    sA[k][t] = ct[(((size_t)b * NSPAN + sid(i, kk, 1)) * Tn + t) * Vn + d];
    sB[k][t] = ct[(((size_t)b * NSPAN + sid(kk + 1, j, 0)) * Tn + t) * Vn + (1 - d)];
  }
  __syncthreads();
  if (tid < Tn) {
    float m = sA[0][tid];
    for (int k = 1; k < K; ++k) m = fmaxf(m, sA[k][tid]);
    mA[tid] = m;
  } else if (tid < 2 * Tn) {
    const int t = tid - Tn;
    float m = sB[0][t];
    for (int k = 1; k < K; ++k) m = fmaxf(m, sB[k][t]);
    mB[t] = m;
  }
  __syncthreads();
  // exponentiate in place; zero-fill pad rows K..KMAX (contribute 0 in exp-space)
  for (int idx = tid; idx < KMAX * Tn; idx += 128) {
    const int k = idx >> 5, t = idx & 31;
    sA[k][t] = (k < K) ? __expf(sA[k][t] - mA[t]) : 0.0f;
    sB[k][t] = (k < K) ? __expf(sB[k][t] - mB[t]) : 0.0f;
  }
  __syncthreads();

  // wave w handles 16x16 tile (t1h, t2h) of the 32x32 (t1,t2) chart
  const int wave = tid >> 5, lane = tid & 31;
  const int t1h = (wave >> 1) * 16, t2h = (wave & 1) * 16;
  const int half = lane >> 4, lm = lane & 15;

  // Preload all fragments (compile-time trip count -> loads hoisted, one wait)
  v2f Af[KMAX / 4], Bf[KMAX / 4];
#pragma unroll
  for (int c = 0; c < KMAX / 4; ++c) {
    // A 16x4 (MxK): lanes 0-15 -> K={4c,4c+1}, lanes 16-31 -> K={4c+2,4c+3}
    Af[c].x = sA[4 * c + 2 * half + 0][t1h + lm];
    Af[c].y = sA[4 * c + 2 * half + 1][t1h + lm];
    Bf[c].x = sB[4 * c + 2 * half + 0][t2h + lm];
    Bf[c].y = sB[4 * c + 2 * half + 1][t2h + lm];
  }
  v8f acc = {};
#pragma unroll
  for (int c = 0; c < KMAX / 4; ++c) {
    acc = __builtin_amdgcn_wmma_f32_16x16x4_f32(
        false, Af[c], false, Bf[c], (short)0, acc, false, false);
  }

  const float* att = (d == 0)
      ? (right_score + ((size_t)b * Sn + j) * Tn * Tn)
      : (left_score + ((size_t)b * Sn + i) * Tn * Tn);
  float* itp = it + ((size_t)b * NSPAN + sid(i, j, d)) * (size_t)(Tn * Tn * Vn);

#pragma unroll
  for (int g = 0; g < 8; ++g) {
    const int t1 = t1h + half * 8 + g; // D layout: VGPR g -> M = g | g+8
    const int t2 = t2h + lm;
    float dd0, dd1;
    if (d == 0) { // dec[b, j, t2, 0, v, 1]
      const float* dp =
          dec + (((((size_t)b * Sn + j) * Tn + t2) * 2 + 0) * Vn + 0) * 2 + 1;
      dd0 = dp[0]; dd1 = dp[2];
    } else {      // dec[b, i, t1, 1, v, 1]
      const float* dp =
          dec + (((((size_t)b * Sn + i) * Tn + t1) * 2 + 1) * Vn + 0) * 2 + 1;
      dd0 = dp[0]; dd1 = dp[2];
    }
    const float base = mA[t1] + mB[t2] + __logf(acc[g]) + att[t1 * Tn + t2];
    itp[(t1 * Tn + t2) * Vn + 0] = base + dd0;
    itp[(t1 * Tn + t2) * Vn + 1] = base + dd1;
  }
}

// Complete spans of length L (exact two-pass logsumexp over K*T terms):
//  d==0: ct[i,j,0][t,v] = lse_{k,tt}( ct[i,i+k,0][tt,0] + it[i+k,j,0][tt,t,v] )
//  d==1: ct[i,j,1][t,v] = lse_{k,tt}( it[i,i+1+k,1][t,tt,v] + ct[i+1+k,j,1][tt,0] )
__global__ __launch_bounds__(64) void comp_kernel(
    float* __restrict__ ct, const float* __restrict__ it, int L) {
  const int span = blockIdx.x;
  const int b = blockIdx.y;
  const int nsp = Sn - L;
  const int d = (span >= nsp) ? 1 : 0;
  const int i = d ? (span - nsp) : span;
  const int j = i + L;
  const int K = L;

  __shared__ float sw[KMAX][Tn];

  const int tid = threadIdx.x;
  for (int idx = tid; idx < K * Tn; idx += 64) {
    const int k = idx >> 5, tt = idx & 31;
    const int sidw = d ? sid(i + 1 + k, j, 1) : sid(i, i + k, 0);
    sw[k][tt] = ct[(((size_t)b * NSPAN + sidw) * Tn + tt) * Vn + 0];
  }
  __syncthreads();

  const int t = tid >> 1, v = tid & 1;

  float m = -3.0e38f;
  for (int k = 0; k < K; ++k) {
    const int sp = d ? sid(i, i + 1 + k, 1) : sid(i + k, j, 0);
    const float* ip = it + ((size_t)b * NSPAN + sp) * (size_t)(Tn * Tn * Vn);
    for (int tt = 0; tt < Tn; ++tt) {
      const float x = sw[k][tt] +
          (d ? ip[(t * Tn + tt) * Vn + v] : ip[(tt * Tn + t) * Vn + v]);
      m = fmaxf(m, x);
    }
  }
  float s = 0.0f;
  for (int k = 0; k < K; ++k) {
    const int sp = d ? sid(i, i + 1 + k, 1) : sid(i + k, j, 0);
    const float* ip = it + ((size_t)b * NSPAN + sp) * (size_t)(Tn * Tn * Vn);
    for (int tt = 0; tt < Tn; ++tt) {
      const float x = sw[k][tt] +
          (d ? ip[(t * Tn + tt) * Vn + v] : ip[(tt * Tn + t) * Vn + v]);
      s += __expf(x - m);
    }
  }
  ct[(((size_t)b * NSPAN + sid(i, j, d)) * Tn + t) * Vn + v] = m + __logf(s);
}

__global__ void final_kernel(const float* __restrict__ ct,
                             const int* __restrict__ sent_lens,
                             float* __restrict__ out) {
  const int b = threadIdx.x;
  if (b >= Bn) return;
  const int len = sent_lens[b];
  out[b] = ct[(((size_t)b * NSPAN + sid(0, len - 1, 1)) * Tn + 0) * Vn + 0];
}

extern "C" void kernel_launch(void* const* d_in, const int* in_sizes, int n_in,
                              void* d_out, int out_size, void* d_ws,
                              size_t ws_size, hipStream_t stream) {
  const float* left  = (const float*)d_in[0];  // (B,S,T,T)
  const float* right = (const float*)d_in[1];  // (B,S,T,T)
  const float* dec   = (const float*)d_in[2];  // (B,S,T,2,V,2)
  const float* unary = (const float*)d_in[3];  // (B,S,T)
  const int*   lens  = (const int*)d_in[4];    // (B,)
  float* out = (float*)d_out;

  float* ct = (float*)d_ws;
  float* it = ct + (size_t)Bn * NSPAN * Tn * Vn; // needs ~108MB total ws

  const int ct_elems = Bn * NSPAN * Tn * Vn;
  fill_ct_kernel<<<(ct_elems + 255) / 256, 256, 0, stream>>>(ct);
  const int diag_elems = Bn * Sn * Tn * Vn * 2;
  init_diag_kernel<<<(diag_elems + 255) / 256, 256, 0, stream>>>(ct, dec, unary);

  for (int L = 1; L < Sn; ++L) {
    dim3 grid(2 * (Sn - L), Bn);
    inc_kernel<<<grid, 128, 0, stream>>>(left, right, dec, ct, it, L);
    comp_kernel<<<grid, 64, 0, stream>>>(ct, it, L);
  }
  final_kernel<<<1, 32, 0, stream>>>(ct, lens, out);
}